// SSMLayer_38053410242661
// MI455X (gfx1250) — compile-verified
//
#include <hip/hip_runtime.h>

// SSM forward for MI455X (gfx1250, wave32, WMMA + TDM).
//   out[b,d,l] = sum_{c,n,tau<=l} C[d,n] B_hat[n,c] Re(z_n^tau) x[b,c,l-tau]
// GEMM -> 1-pole complex scan -> GEMM, 3-phase chunked scan, TDM-staged x tiles.

#define BATCHN 8
#define CIN    64
#define COUT   64
#define NCO    256         // coeffs
#define TLEN   16384
#define LCO    128         // outer scan chunk (ws state granularity)
#define SUB    32          // inner LDS staging sub-chunk
#define NSUB   (LCO/SUB)   // 4
#define NCH    (TLEN/LCO)  // 128 chunks
#define XSTR   (SUB+4)     // 36: padded LDS row stride for x tile
#define HSTR   (SUB+1)     // 33: odd stride -> conflict-free scan reads

// workspace layout (float offsets); total ~4.3 MB
#define WS_ZR    0
#define WS_ZI    256
#define WS_ZLR   512
#define WS_ZLI   768
#define WS_BHAT  1024
#define WS_S     (WS_BHAT + NCO*CIN)
#define WS_CARRY (WS_S + BATCHN*NCH*NCO*2)

typedef __attribute__((ext_vector_type(2))) float        v2f;
typedef __attribute__((ext_vector_type(8))) float        v8f;
typedef __attribute__((ext_vector_type(4))) unsigned int v4u;
typedef __attribute__((ext_vector_type(8))) int          v8i;
typedef __attribute__((ext_vector_type(4))) int          v4i;

// ---------------------------------------------------------------------------
// TDM: DMA a 64(row) x 32(float) tile, global row stride TLEN floats, into LDS
// with 16B padding after each 128B row (-> XSTR=36-float LDS rows).
// D# per ISA 8.3/8.4: group0 = {count=1, lds_addr, global_addr, type=2},
// group1 = {data_size=4B, pad_en, pad_interval=32dw, pad_amount=4dw,
//           tensor_dim0=TLEN, tensor_dim1=CIN, tile_dim0=SUB, tile_dim1=CIN,
//           tensor_dim0_stride=TLEN}; groups 2/3 zero (2D tensor).
// This toolchain exposes the 6-arg builtin (extra int32x8 group, then cpol).
// ---------------------------------------------------------------------------
__device__ inline void tdm_issue_x(const float* gsrc, const float* ldst) {
  const unsigned long long ga = (unsigned long long)(uintptr_t)gsrc;
  const unsigned int lds = (unsigned int)(uintptr_t)ldst;   // low 32b = LDS byte addr
  v4u g0;
  g0[0] = 1u;                                               // count=1, user mode
  g0[1] = lds;
  g0[2] = (unsigned int)ga;
  g0[3] = (unsigned int)((ga >> 32) & 0x01FFFFFFu) | (2u << 30);  // type=2 (image)
  v8i g1;
  g1[0] = (2 << 16)          // data_size = 4 bytes
        | (1 << 20)          // pad_enable
        | (4 << 22)          // pad_interval: 2<<4 = 32 DWORDs (one 128B row)
        | (3 << 25);         // pad_amount: 4 DWORDs (16B) -> row stride 144B
  g1[1] = (int)((TLEN & 0xFFFF) << 16);                 // tensor_dim0[15:0]
  g1[2] = (int)((TLEN >> 16) | ((CIN & 0xFFFF) << 16)); // td0[31:16] | td1[15:0]
  g1[3] = (int)((SUB & 0xFFFF) << 16);                  // td1[31:16] | tile_dim0
  g1[4] = CIN;                                          // tile_dim1 | tile_dim2=0
  g1[5] = TLEN;                                         // tensor_dim0_stride[31:0]
  g1[6] = 0;                                            // stride0[47:32] | stride1 lo
  g1[7] = 0;
  const v4i g2 = {0, 0, 0, 0};
  const v4i g3 = {0, 0, 0, 0};
  const v8i g4 = {0, 0, 0, 0, 0, 0, 0, 0};
  __builtin_amdgcn_tensor_load_to_lds(g0, g1, g2, g3, g4, 0);
}

// ---------------------------------------------------------------------------
// Precompute z_n = exp(dtA), z_n^LCO, and B_hat = B * dt.
// ---------------------------------------------------------------------------
__global__ __launch_bounds__(256) void ssm_params(const float* __restrict__ A,
                                                  const float* __restrict__ Bm,
                                                  const float* __restrict__ logdt,
                                                  float* __restrict__ ws) {
  const int n = threadIdx.x;                 // 0..255
  const float lar = A[2 * n];
  const float aim = A[2 * n + 1];
  const float dt  = expf(logdt[n]);
  const float sp  = (lar > 20.f) ? lar : log1pf(expf(lar));   // softplus
  const float a   = -dt * sp;
  const float w   =  dt * aim;
  const float ea  = expf(a);
  ws[WS_ZR + n] = ea * cosf(w);
  ws[WS_ZI + n] = ea * sinf(w);
  const float eaL = expf(a * (float)LCO);
  ws[WS_ZLR + n] = eaL * cosf(w * (float)LCO);
  ws[WS_ZLI + n] = eaL * sinf(w * (float)LCO);
#pragma unroll 8
  for (int c = 0; c < CIN; ++c)
    ws[WS_BHAT + n * CIN + c] = Bm[n * CIN + c] * dt;
}

// ---------------------------------------------------------------------------
// h[n0:n0+16, l0:l0+16] tiles via V_WMMA_F32_16X16X4_F32, K-chained over c.
// Constant trip count so the loop is provably wave-uniform (EXEC all-1s).
// ---------------------------------------------------------------------------
__device__ inline void compute_h_tiles(const float* __restrict__ bhat,
                                       const float* __restrict__ xbuf,
                                       float* __restrict__ hbuf) {
  const int wave = threadIdx.x >> 5;
  const int lane = threadIdx.x & 31;
  const int hi = lane >> 4;      // lane half selects K pair / M half
  const int li = lane & 15;
#pragma unroll
  for (int t = 0; t < (NCO / 16) * (SUB / 16) / 8; ++t) {   // 4 tiles per wave
    const int tile = wave + (t << 3);
    const int n0 = (tile >> 1) << 4;       // SUB/16 == 2 l-tiles
    const int l0 = (tile & 1) << 4;
    const float* brow = bhat + (n0 + li) * CIN + 2 * hi;
    v8f acc = {};
#pragma unroll 4
    for (int c0 = 0; c0 < CIN; c0 += 4) {
      v2f av, bv;
      av.x = brow[c0];
      av.y = brow[c0 + 1];
      bv.x = xbuf[(c0 + hi) * XSTR + l0 + li];
      bv.y = xbuf[(c0 + 2 + hi) * XSTR + l0 + li];
      acc = __builtin_amdgcn_wmma_f32_16x16x4_f32(false, av, false, bv,
                                                  (short)0, acc, false, false);
    }
    float* hp = hbuf + l0 + li;
#pragma unroll
    for (int r = 0; r < 8; ++r)
      hp[(n0 + r + 8 * hi) * HSTR] = acc[r];
  }
}

// ---------------------------------------------------------------------------
// Phase 1: per (b, chunk) compute local chunk-end state S (scan seeded at 0).
// Software pipeline: TDM(s+1) overlaps compute(s); s_wait_tensorcnt before the
// buffer-swap barrier.
// ---------------------------------------------------------------------------
__global__ __launch_bounds__(256) void ssm_chunk_state(const float* __restrict__ x,
                                                       float* __restrict__ ws) {
  __shared__ __align__(16) float xbuf[2][CIN * XSTR];
  __shared__ float hbuf[NCO * HSTR];
  const int ch = blockIdx.x;
  const int b  = blockIdx.y;
  const int n  = threadIdx.x;              // thread n owns coefficient n
  const bool wave0 = (threadIdx.x >> 5) == 0;
  const float zr = ws[WS_ZR + n], zi = ws[WS_ZI + n];
  float sr = 0.f, si = 0.f;
  const float* xg = x + ((long)b * CIN) * TLEN + (long)ch * LCO;

  if (wave0) {
    tdm_issue_x(xg, xbuf[0]);
    __builtin_amdgcn_s_wait_tensorcnt(0);
  }
  __syncthreads();

  for (int s = 0; s < NSUB; ++s) {
    if (wave0 && (s + 1 < NSUB))
      tdm_issue_x(xg + (s + 1) * SUB, xbuf[(s + 1) & 1]);   // prefetch next tile
    compute_h_tiles(ws + WS_BHAT, xbuf[s & 1], hbuf);
    __syncthreads();                       // hbuf ready
    const float* hrow = hbuf + n * HSTR;
#pragma unroll 8
    for (int l = 0; l < SUB; ++l) {        // s = z*s + h
      const float h  = hrow[l];
      const float nr = fmaf(zr, sr, fmaf(-zi, si, h));
      const float ni = fmaf(zr, si, zi * sr);
      sr = nr; si = ni;
    }
    if (wave0 && (s + 1 < NSUB))
      __builtin_amdgcn_s_wait_tensorcnt(0);
    __syncthreads();                       // xbuf[(s+1)&1] ready, hbuf reusable
  }
  const long idx = (((long)b * NCH + ch) * NCO + n) * 2;
  ws[WS_S + idx]     = sr;
  ws[WS_S + idx + 1] = si;
}

// ---------------------------------------------------------------------------
// Phase 2: prefix scan over chunk states: carry[j] = z^LCO*carry[j-1] + S[j-1].
// ---------------------------------------------------------------------------
__global__ __launch_bounds__(256) void ssm_scan(float* __restrict__ ws) {
  const int b = blockIdx.x;
  const int n = threadIdx.x;
  const float zlr = ws[WS_ZLR + n], zli = ws[WS_ZLI + n];
  float cr = 0.f, ci = 0.f;
  for (int j = 0; j < NCH; ++j) {
    const long idx = (((long)b * NCH + j) * NCO + n) * 2;
    ws[WS_CARRY + idx]     = cr;
    ws[WS_CARRY + idx + 1] = ci;
    const float srj = ws[WS_S + idx], sij = ws[WS_S + idx + 1];
    const float nr = fmaf(zlr, cr, fmaf(-zli, ci, srj));
    const float ni = fmaf(zlr, ci, fmaf(zli, cr, sij));
    cr = nr; ci = ni;
  }
}

// ---------------------------------------------------------------------------
// Phase 3: seeded replay + output projection out = C @ y via WMMA.
// ---------------------------------------------------------------------------
__global__ __launch_bounds__(256) void ssm_chunk_out(const float* __restrict__ x,
                                                     const float* __restrict__ Cm,
                                                     float* __restrict__ outp,
                                                     float* __restrict__ ws) {
  __shared__ __align__(16) float xbuf[2][CIN * XSTR];
  __shared__ float hbuf[NCO * HSTR];
  const int ch = blockIdx.x;
  const int b  = blockIdx.y;
  const int n  = threadIdx.x;
  const int wave = threadIdx.x >> 5;
  const int lane = threadIdx.x & 31;
  const int hi = lane >> 4, li = lane & 15;
  const bool wave0 = (wave == 0);
  const float zr = ws[WS_ZR + n], zi = ws[WS_ZI + n];
  const long cidx = (((long)b * NCH + ch) * NCO + n) * 2;
  float sr = ws[WS_CARRY + cidx];
  float si = ws[WS_CARRY + cidx + 1];
  const float* xg = x + ((long)b * CIN) * TLEN + (long)ch * LCO;

  if (wave0) {
    tdm_issue_x(xg, xbuf[0]);
    __builtin_amdgcn_s_wait_tensorcnt(0);
  }
  __syncthreads();

  for (int s = 0; s < NSUB; ++s) {
    if (wave0 && (s + 1 < NSUB))
      tdm_issue_x(xg + (s + 1) * SUB, xbuf[(s + 1) & 1]);
    compute_h_tiles(ws + WS_BHAT, xbuf[s & 1], hbuf);
    __syncthreads();                       // hbuf(h) ready
    float* hrow = hbuf + n * HSTR;
#pragma unroll 8
    for (int l = 0; l < SUB; ++l) {        // y[l] = Re(s), stored in-place
      const float h  = hrow[l];
      const float nr = fmaf(zr, sr, fmaf(-zi, si, h));
      const float ni = fmaf(zr, si, zi * sr);
      sr = nr; si = ni;
      hrow[l] = sr;
    }
    __syncthreads();                       // hbuf(y) ready
    // out[d0:d0+16, l0:l0+16] = C[16d x 256n] @ y[256n x 16l]; 8 tiles, 1/wave
    {
      const int d0 = (wave >> 1) << 4;     // SUB/16 == 2 l-tiles
      const int l0 = (wave & 1) << 4;
      const float* crow = Cm + (d0 + li) * NCO + 2 * hi;
      v8f acc = {};
#pragma unroll 4
      for (int n0 = 0; n0 < NCO; n0 += 4) {
        v2f av, bv;
        av.x = crow[n0];
        av.y = crow[n0 + 1];
        bv.x = hbuf[(n0 + hi) * HSTR + l0 + li];
        bv.y = hbuf[(n0 + 2 + hi) * HSTR + l0 + li];
        acc = __builtin_amdgcn_wmma_f32_16x16x4_f32(false, av, false, bv,
                                                    (short)0, acc, false, false);
      }
      float* op = outp + ((long)b * COUT + d0 + 8 * hi) * TLEN +
                  (long)ch * LCO + s * SUB + l0 + li;
#pragma unroll
      for (int r = 0; r < 8; ++r)
        op[(long)r * TLEN] = acc[r];
    }
    if (wave0 && (s + 1 < NSUB))
      __builtin_amdgcn_s_wait_tensorcnt(0);
    __syncthreads();                       // next xbuf ready, hbuf reusable
  }
}

// ---------------------------------------------------------------------------
extern "C" void kernel_launch(void* const* d_in, const int* in_sizes, int n_in,
                              void* d_out, int out_size, void* d_ws, size_t ws_size,
                              hipStream_t stream) {
  (void)in_sizes; (void)n_in; (void)out_size; (void)ws_size;
  const float* x     = (const float*)d_in[0];   // [8,64,16384]
  const float* A     = (const float*)d_in[1];   // [256,2]
  const float* Bm    = (const float*)d_in[2];   // [256,64]
  const float* logdt = (const float*)d_in[3];   // [256]
  const float* Cm    = (const float*)d_in[4];   // [64,256]
  float* outp = (float*)d_out;                  // [8,64,16384]
  float* ws   = (float*)d_ws;

  ssm_params<<<1, 256, 0, stream>>>(A, Bm, logdt, ws);
  ssm_chunk_state<<<dim3(NCH, BATCHN), 256, 0, stream>>>(x, ws);
  ssm_scan<<<BATCHN, 256, 0, stream>>>(ws);
  ssm_chunk_out<<<dim3(NCH, BATCHN), 256, 0, stream>>>(x, Cm, outp, ws);
}